// Sent_Posit_Drmm_Modeler_73443940761857
// MI455X (gfx1250) — compile-verified
//
#include <hip/hip_runtime.h>
#include <cstddef>

// ---------------------------------------------------------------------------
// Sent_Posit_Drmm_Modeler on MI455X (gfx1250, wave32, WMMA f32 16x16x4)
// Fused similarity-GEMM + top-5 pooling; convs expressed as WMMA GEMMs with
// dual independent accumulator chains per wave.
// Workspace requirement: ~61 MB of f32 scratch in d_ws.
// ---------------------------------------------------------------------------

typedef __attribute__((ext_vector_type(2))) float v2f;
typedef __attribute__((ext_vector_type(8))) float v8f;

#define QLEN 256
#define DLEN 8192
#define EMB  300
#define SROW 304   // padded row stride (multiple of 16)

__device__ __forceinline__ v8f wmma4(v2f a, v2f b, v8f c) {
  // D = A(16x4 f32) * B(4x16 f32) + C(16x16 f32)
  return __builtin_amdgcn_wmma_f32_16x16x4_f32(
      false, a, false, b, (short)0, c, false, false);
}

__device__ __forceinline__ void top5_insert(float (&t)[5], float v) {
  if (v > t[4]) {
    t[4] = v;
#pragma unroll
    for (int i = 4; i > 0; --i) {
      if (t[i] > t[i - 1]) { float tmp = t[i - 1]; t[i - 1] = t[i]; t[i] = tmp; }
      else break;
    }
  }
}

// ---------------------------------------------------------------------------
// K1: gather embedding rows into padded buffer + mask floats
// ---------------------------------------------------------------------------
__global__ void gather_embed(const int* __restrict__ tokens,
                             const float* __restrict__ embeds,
                             float* __restrict__ out,
                             float* __restrict__ maskf, int L) {
  int row = blockIdx.x;
  if (row >= L) return;
  int tok = tokens[row];
  const float* src = embeds + (size_t)tok * EMB;
  float* dst = out + (size_t)row * SROW;
  for (int c = threadIdx.x; c < SROW; c += blockDim.x)
    dst[c] = (c < EMB) ? src[c] : 0.0f;
  if (threadIdx.x == 0) maskf[row] = (tok > 1) ? 1.0f : 0.0f;
}

// ---------------------------------------------------------------------------
// K2: 1/||row||2 over first 300 columns
// ---------------------------------------------------------------------------
__global__ void row_invnorm(const float* __restrict__ x, float* __restrict__ invn) {
  __shared__ float red[128];
  int row = blockIdx.x;
  const float* p = x + (size_t)row * SROW;
  float s = 0.0f;
  for (int c = threadIdx.x; c < EMB; c += blockDim.x) { float v = p[c]; s += v * v; }
  red[threadIdx.x] = s;
  __syncthreads();
  for (int off = 64; off > 0; off >>= 1) {
    if ((int)threadIdx.x < off) red[threadIdx.x] += red[threadIdx.x + off];
    __syncthreads();
  }
  if (threadIdx.x == 0) invn[row] = rsqrtf(fmaxf(red[0], 1e-30f));
}

// ---------------------------------------------------------------------------
// K3: conv-as-GEMM with WMMA, PReLU + residual epilogue.
//   y[t,o] = sum_j sum_e W[o,0,j,e] * x[t+off0+j, e];  out = prelu(y)+x
// One wave computes a 16x32 output tile: two independent WMMA chains share
// one A operand (halves A traffic, hides WMMA latency).
// grid = (L/16, ceil(320/32)=10)
// ---------------------------------------------------------------------------
__global__ void __launch_bounds__(32)
conv_wmma(const float* __restrict__ x, const float* __restrict__ W,
          const float* __restrict__ alpha, float* __restrict__ out,
          int L, int fs, int off0) {
  int lane = threadIdx.x & 31;
  int half = lane >> 4;
  int lm   = lane & 15;
  int t0   = blockIdx.x * 16;
  int o0   = blockIdx.y * 32 + lm;       // channel tile 0
  int o1   = o0 + 16;                    // channel tile 1
  bool bv0 = (o0 < EMB), bv1 = (o1 < EMB);
  float bs0 = bv0 ? 1.0f : 0.0f, bs1 = bv1 ? 1.0f : 0.0f;
  int oc0 = bv0 ? o0 : (EMB - 1), oc1 = bv1 ? o1 : (EMB - 1);

  v8f c0 = {}, c1 = {};
  for (int j = 0; j < fs; ++j) {
    int r = t0 + lm + off0 + j;
    bool avalid = (r < L);
    float as = avalid ? 1.0f : 0.0f;
    const float* ap  = x + (size_t)(avalid ? r : 0) * SROW + half * 2;
    const float* bp0 = W + ((size_t)oc0 * fs + j) * EMB + half * 2;
    const float* bp1 = W + ((size_t)oc1 * fs + j) * EMB + half * 2;
#pragma unroll 5
    for (int k = 0; k < EMB; k += 4) {
      v2f a, b0, b1v;
      a.x   = as  * ap[k];  a.y   = as  * ap[k + 1];
      b0.x  = bs0 * bp0[k]; b0.y  = bs0 * bp0[k + 1];
      b1v.x = bs1 * bp1[k]; b1v.y = bs1 * bp1[k + 1];
      c0 = wmma4(a, b0,  c0);
      c1 = wmma4(a, b1v, c1);
    }
  }
  float al = alpha[0];
#pragma unroll
  for (int i = 0; i < 8; ++i) {
    int row  = t0 + i + 8 * half;
    int col0 = blockIdx.y * 32 + lm;
    int col1 = col0 + 16;
    float y0 = c0[i]; y0 = (y0 >= 0.0f) ? y0 : al * y0;
    float y1 = c1[i]; y1 = (y1 >= 0.0f) ? y1 : al * y1;
    if (col0 < SROW) {
      size_t idx = (size_t)row * SROW + col0;
      out[idx] = bv0 ? (y0 + x[idx]) : 0.0f;
    }
    if (col1 < SROW) {
      size_t idx = (size_t)row * SROW + col1;
      out[idx] = bv1 ? (y1 + x[idx]) : 0.0f;
    }
  }
}

// ---------------------------------------------------------------------------
// K4: fused cosine-similarity (3 GEMMs, S^T = D * Q^T) + top-5 pooling for
// 4 trackers (doc_sim.T / ins / bi / tri) + 8->8->1 PReLU MLP.
// grid.x = QLEN/16 query tiles; 4 waves per block stride over doc tiles.
// ---------------------------------------------------------------------------
__global__ void __launch_bounds__(128)
sim_topk(const float* __restrict__ qe,  const float* __restrict__ qbi,
         const float* __restrict__ qtr,
         const float* __restrict__ de,  const float* __restrict__ dbi,
         const float* __restrict__ dtr,
         const float* __restrict__ qn_e, const float* __restrict__ qn_bi,
         const float* __restrict__ qn_tr,
         const float* __restrict__ dn_e, const float* __restrict__ dn_bi,
         const float* __restrict__ dn_tr,
         const float* __restrict__ qmaskf, const float* __restrict__ dmaskf,
         const float* __restrict__ doc_sim,   // [DLEN][QLEN]
         const float* __restrict__ w1, const float* __restrict__ b1,
         const float* __restrict__ w2, const float* __restrict__ b2,
         const float* __restrict__ a1p, const float* __restrict__ a2p,
         float* __restrict__ lo_out) {
  __shared__ float ldsq[3][16][305];          // stride 305 -> conflict-free
  __shared__ float topbuf[4][32][4][5];

  int tid  = threadIdx.x;
  int wave = tid >> 5;
  int lane = tid & 31;
  int half = lane >> 4;
  int lm   = lane & 15;
  int qt0  = blockIdx.x * 16;

  // Stage the 16-query tile (e / bi / tri) in LDS once.
  for (int idx = tid; idx < 16 * EMB; idx += 128) {
    int r = idx / EMB, cc = idx % EMB;
    size_t g = (size_t)(qt0 + r) * SROW + cc;
    ldsq[0][r][cc] = qe[g];
    ldsq[1][r][cc] = qbi[g];
    ldsq[2][r][cc] = qtr[g];
  }
  __syncthreads();

  int q = qt0 + lm;
  float qne = qn_e[q], qnb = qn_bi[q], qnt = qn_tr[q], qm = qmaskf[q];

  float t_oh[5], t_in[5], t_bi[5], t_tr[5];
#pragma unroll
  for (int i = 0; i < 5; ++i) { t_oh[i] = t_in[i] = t_bi[i] = t_tr[i] = -1e30f; }

  for (int tile = wave; tile < DLEN / 16; tile += 4) {
    int dr = tile * 16 + lm;
    const float* ae  = de  + (size_t)dr * SROW + half * 2;
    const float* ab  = dbi + (size_t)dr * SROW + half * 2;
    const float* at  = dtr + (size_t)dr * SROW + half * 2;
    const float* b0  = &ldsq[0][lm][half * 2];
    const float* b1q = &ldsq[1][lm][half * 2];
    const float* b2q = &ldsq[2][lm][half * 2];
    v8f ce = {}, cb = {}, ct = {};
#pragma unroll 5
    for (int k = 0; k < EMB; k += 4) {
      v2f a, b;
      a.x = ae[k]; a.y = ae[k + 1]; b.x = b0[k];  b.y = b0[k + 1];
      ce = wmma4(a, b, ce);
      a.x = ab[k]; a.y = ab[k + 1]; b.x = b1q[k]; b.y = b1q[k + 1];
      cb = wmma4(a, b, cb);
      a.x = at[k]; a.y = at[k + 1]; b.x = b2q[k]; b.y = b2q[k + 1];
      ct = wmma4(a, b, ct);
    }
#pragma unroll
    for (int i = 0; i < 8; ++i) {
      int drow = tile * 16 + i + 8 * half;
      float mk = qm * dmaskf[drow];
      top5_insert(t_in, ce[i] * qne * dn_e[drow] * mk);
      top5_insert(t_bi, cb[i] * qnb * dn_bi[drow]);
      top5_insert(t_tr, ct[i] * qnt * dn_tr[drow]);
      top5_insert(t_oh, doc_sim[(size_t)drow * QLEN + q]);
    }
  }

#pragma unroll
  for (int i = 0; i < 5; ++i) {
    topbuf[wave][lane][0][i] = t_oh[i];
    topbuf[wave][lane][1][i] = t_in[i];
    topbuf[wave][lane][2][i] = t_bi[i];
    topbuf[wave][lane][3][i] = t_tr[i];
  }
  __syncthreads();

  if (tid < 16) {
    float feat[8];
    for (int m = 0; m < 4; ++m) {
      float fin[5];
#pragma unroll
      for (int i = 0; i < 5; ++i) fin[i] = -1e30f;
      for (int w = 0; w < 4; ++w)
        for (int h = 0; h < 2; ++h)
          for (int i = 0; i < 5; ++i)
            top5_insert(fin, topbuf[w][tid + 16 * h][m][i]);
      float s = fin[0] + fin[1] + fin[2] + fin[3] + fin[4];
      feat[2 * m]     = fin[0];      // max
      feat[2 * m + 1] = s * 0.2f;    // avg of top-5
    }
    float a1 = a1p[0], a2 = a2p[0];
    float l1v[8];
    for (int j = 0; j < 8; ++j) {
      float s = b1[j];
      for (int k2 = 0; k2 < 8; ++k2) s += feat[k2] * w1[j * 8 + k2];
      l1v[j] = (s >= 0.0f) ? s : a1 * s;
    }
    float s2 = b2[0];
    for (int j = 0; j < 8; ++j) s2 += l1v[j] * w2[j];
    s2 = (s2 >= 0.0f) ? s2 : a2 * s2;
    lo_out[qt0 + tid] = s2;
  }
}

// ---------------------------------------------------------------------------
// K5: mean over queries + margin loss, 5 scalar outputs
// ---------------------------------------------------------------------------
__global__ void finalize(const float* __restrict__ lo1,
                         const float* __restrict__ lo2,
                         float* __restrict__ out) {
  __shared__ float s1[256], s2[256];
  int t = threadIdx.x;
  s1[t] = lo1[t];
  s2[t] = lo2[t];
  __syncthreads();
  for (int off = 128; off > 0; off >>= 1) {
    if (t < off) { s1[t] += s1[t + off]; s2[t] += s2[t + off]; }
    __syncthreads();
  }
  if (t == 0) {
    float d1 = s1[0] * (1.0f / 256.0f);
    float d2 = s2[0] * (1.0f / 256.0f);
    float loss = fmaxf(0.0f, -(d1 - d2) + 0.9f);
    out[0] = loss; out[1] = d1; out[2] = d2; out[3] = loss; out[4] = 0.0f;
  }
}

// ---------------------------------------------------------------------------
extern "C" void kernel_launch(void* const* d_in, const int* in_sizes, int n_in,
                              void* d_out, int out_size, void* d_ws, size_t ws_size,
                              hipStream_t stream) {
  (void)in_sizes; (void)n_in; (void)out_size; (void)ws_size;
  const int*   doc1     = (const int*)d_in[0];
  const int*   doc2     = (const int*)d_in[1];
  const int*   question = (const int*)d_in[2];
  const float* doc1_sim = (const float*)d_in[3];
  const float* doc2_sim = (const float*)d_in[4];
  const float* embeds   = (const float*)d_in[5];
  const float* filt_bi  = (const float*)d_in[6];   // [300][1][2][300]
  const float* filt_tri = (const float*)d_in[7];   // [300][1][3][300]
  const float* a_bi     = (const float*)d_in[8];
  const float* a_tri    = (const float*)d_in[9];
  const float* w1       = (const float*)d_in[10];
  const float* b1       = (const float*)d_in[11];
  const float* w2       = (const float*)d_in[12];
  const float* b2       = (const float*)d_in[13];
  const float* a1       = (const float*)d_in[14];
  const float* a2       = (const float*)d_in[15];

  float* ws = (float*)d_ws;
  size_t off = 0;
  auto alloc = [&](size_t n) { float* p = ws + off; off += n; return p; };

  float* qe    = alloc((size_t)QLEN * SROW);
  float* qbi   = alloc((size_t)QLEN * SROW);
  float* qtr   = alloc((size_t)QLEN * SROW);
  float* d1e   = alloc((size_t)DLEN * SROW);
  float* d1bi  = alloc((size_t)DLEN * SROW);
  float* d1tr  = alloc((size_t)DLEN * SROW);
  float* d2e   = alloc((size_t)DLEN * SROW);
  float* d2bi  = alloc((size_t)DLEN * SROW);
  float* d2tr  = alloc((size_t)DLEN * SROW);
  float* qn_e  = alloc(QLEN); float* qn_bi = alloc(QLEN); float* qn_tr = alloc(QLEN);
  float* d1n_e = alloc(DLEN); float* d1n_bi = alloc(DLEN); float* d1n_tr = alloc(DLEN);
  float* d2n_e = alloc(DLEN); float* d2n_bi = alloc(DLEN); float* d2n_tr = alloc(DLEN);
  float* qmk   = alloc(QLEN);
  float* d1mk  = alloc(DLEN);
  float* d2mk  = alloc(DLEN);
  float* lo1   = alloc(QLEN);
  float* lo2   = alloc(QLEN);

  // 1) gather + masks
  gather_embed<<<QLEN, 128, 0, stream>>>(question, embeds, qe, qmk, QLEN);
  gather_embed<<<DLEN, 128, 0, stream>>>(doc1, embeds, d1e, d1mk, DLEN);
  gather_embed<<<DLEN, 128, 0, stream>>>(doc2, embeds, d2e, d2mk, DLEN);

  // 2) convs (WMMA GEMMs): bi -> rows t+1,t+2 (off0=1); tri -> t..t+2 (off0=0)
  dim3 cq(QLEN / 16, 10), cd(DLEN / 16, 10);
  conv_wmma<<<cq, 32, 0, stream>>>(qe,  filt_bi,  a_bi,  qbi,  QLEN, 2, 1);
  conv_wmma<<<cq, 32, 0, stream>>>(qe,  filt_tri, a_tri, qtr,  QLEN, 3, 0);
  conv_wmma<<<cd, 32, 0, stream>>>(d1e, filt_bi,  a_bi,  d1bi, DLEN, 2, 1);
  conv_wmma<<<cd, 32, 0, stream>>>(d1e, filt_tri, a_tri, d1tr, DLEN, 3, 0);
  conv_wmma<<<cd, 32, 0, stream>>>(d2e, filt_bi,  a_bi,  d2bi, DLEN, 2, 1);
  conv_wmma<<<cd, 32, 0, stream>>>(d2e, filt_tri, a_tri, d2tr, DLEN, 3, 0);

  // 3) inverse row norms (cosine denominators)
  row_invnorm<<<QLEN, 128, 0, stream>>>(qe,   qn_e);
  row_invnorm<<<QLEN, 128, 0, stream>>>(qbi,  qn_bi);
  row_invnorm<<<QLEN, 128, 0, stream>>>(qtr,  qn_tr);
  row_invnorm<<<DLEN, 128, 0, stream>>>(d1e,  d1n_e);
  row_invnorm<<<DLEN, 128, 0, stream>>>(d1bi, d1n_bi);
  row_invnorm<<<DLEN, 128, 0, stream>>>(d1tr, d1n_tr);
  row_invnorm<<<DLEN, 128, 0, stream>>>(d2e,  d2n_e);
  row_invnorm<<<DLEN, 128, 0, stream>>>(d2bi, d2n_bi);
  row_invnorm<<<DLEN, 128, 0, stream>>>(d2tr, d2n_tr);

  // 4) fused similarity + top-5 + MLP, one launch per document
  sim_topk<<<QLEN / 16, 128, 0, stream>>>(
      qe, qbi, qtr, d1e, d1bi, d1tr,
      qn_e, qn_bi, qn_tr, d1n_e, d1n_bi, d1n_tr,
      qmk, d1mk, doc1_sim, w1, b1, w2, b2, a1, a2, lo1);
  sim_topk<<<QLEN / 16, 128, 0, stream>>>(
      qe, qbi, qtr, d2e, d2bi, d2tr,
      qn_e, qn_bi, qn_tr, d2n_e, d2n_bi, d2n_tr,
      qmk, d2mk, doc2_sim, w1, b1, w2, b2, a1, a2, lo2);

  // 5) scalar outputs
  finalize<<<1, 256, 0, stream>>>(lo1, lo2, (float*)d_out);
}